// Convolution_74594991997069
// MI455X (gfx1250) — compile-verified
//
#include <hip/hip_runtime.h>

// Rulebook sparse convolution for MI455X (gfx1250, wave32).
// Memory-bound (~1.4-2 GB traffic vs 11 GFLOP) -> keep f32, use
// V_WMMA_F32_16X16X4_F32 for the per-tile 16x32 @ 32x32 GEMMs.

typedef __attribute__((ext_vector_type(2))) float v2f;
typedef __attribute__((ext_vector_type(8))) float v8f;

#define NCIN  32
#define NCOUT 32

// out[i] = bias[i % 32]  (harness poisons d_out; scatter-add lands on top)
__global__ void bias_init_kernel(float* __restrict__ out,
                                 const float* __restrict__ bias,
                                 int total) {
  int i = blockIdx.x * blockDim.x + threadIdx.x;
  if (i < total) out[i] = bias[i & (NCOUT - 1)];
}

__global__ __launch_bounds__(256) void sparse_conv_wmma_f32(
    const float* __restrict__ feat,    // [N_IN, 32]
    const float* __restrict__ weight,  // [K*32, 32]
    const int*   __restrict__ in_idx,  // [K, R]
    const int*   __restrict__ out_idx, // [K, R]
    float* __restrict__ out,           // [N_OUT, 32]
    int R, int groupsPerK) {
  const int lane = threadIdx.x & 31;
  const int wave = threadIdx.x >> 5;          // 8 waves / block
  const int k     = blockIdx.x / groupsPerK;  // filter offset this block owns
  const int group = blockIdx.x % groupsPerK;

  const int half = lane >> 4;                 // 0: lanes 0-15, 1: lanes 16-31
  const int l16  = lane & 15;

  // ---- B fragments for offset k, held in registers across all tiles.
  // B(4x16) layout: v0 = row (4c+2h), v1 = row (4c+1+2h); N = l16 (+16n).
  const float* Wk = weight + (size_t)k * NCIN * NCOUT;
  v2f b[8][2];
#pragma unroll
  for (int c = 0; c < 8; ++c) {
#pragma unroll
    for (int n = 0; n < 2; ++n) {
      const int row0 = 4 * c + 2 * half;
      const int col  = 16 * n + l16;
      b[c][n].x = Wk[(size_t)row0 * NCOUT + col];
      b[c][n].y = Wk[(size_t)(row0 + 1) * NCOUT + col];
    }
  }

  const int numTiles   = (R + 15) >> 4;       // 16 rules per wave-tile
  const int waveStride = groupsPerK * 8;

  for (int tile = group * 8 + wave; tile < numTiles; tile += waveStride) {
    const int base = tile * 16;
    const int rule = base + l16;              // this lane's rule (both halves)

    int iin = 0, oidx = -1;
    if (rule < R) {
      iin  = in_idx [(size_t)k * R + rule];
      oidx = out_idx[(size_t)k * R + rule];
    }

    // ---- Gather: A(16x4) chunks. Lane reads channels (4c + 2*half + {0,1})
    // of row l16 -> each 128B feature row is fully consumed by 2 lanes.
    const float* rowp = feat + (size_t)iin * NCIN + 2 * half;
    v2f a[8];
#pragma unroll
    for (int c = 0; c < 8; ++c) {
      a[c] = *(const v2f*)(rowp + 4 * c);     // global_load_b64
    }

    // ---- 16x32 @ 32x32: 8 K-chunks x 2 N-tiles, two independent acc chains
    // interleaved so WMMA->WMMA RAW hazards overlap.
    v8f acc0 = {};
    v8f acc1 = {};
#pragma unroll
    for (int c = 0; c < 8; ++c) {
      acc0 = __builtin_amdgcn_wmma_f32_16x16x4_f32(
          false, a[c], false, b[c][0], (short)0, acc0, false, false);
      acc1 = __builtin_amdgcn_wmma_f32_16x16x4_f32(
          false, a[c], false, b[c][1], (short)0, acc1, false, false);
    }

    // ---- Scatter-add. D layout: VGPR j -> M = j + 8*half, N = l16 (+16).
    // Route out-row indices to the right lanes via ds_bpermute (no LDS).
#pragma unroll
    for (int j = 0; j < 8; ++j) {
      const int m   = j + 8 * half;
      const int row = __builtin_amdgcn_ds_bpermute(m << 2, oidx);
      if (row >= 0) {
        atomicAdd(out + (size_t)row * NCOUT + l16,      acc0[j]);
        atomicAdd(out + (size_t)row * NCOUT + 16 + l16, acc1[j]);
      }
    }
  }
}

extern "C" void kernel_launch(void* const* d_in, const int* in_sizes, int n_in,
                              void* d_out, int out_size, void* d_ws, size_t ws_size,
                              hipStream_t stream) {
  const float* feat    = (const float*)d_in[0];
  const float* weight  = (const float*)d_in[1];
  const float* bias    = (const float*)d_in[2];
  const int*   in_idx  = (const int*)d_in[3];
  const int*   out_idx = (const int*)d_in[4];
  float*       out     = (float*)d_out;

  const int K = in_sizes[1] / (NCIN * NCOUT);  // weight is [K*32, 32]
  const int R = in_sizes[3] / K;               // in_idx is [K, R]

  // 1) out = broadcast(bias)
  bias_init_kernel<<<(out_size + 255) / 256, 256, 0, stream>>>(out, bias, out_size);

  // 2) gather -> WMMA -> scatter-add. One offset per block; 8 waves/block each
  // stride over 16-rule tiles. 27*128 = 3456 blocks keeps the WGPs saturated
  // while B stays resident in registers per offset.
  const int groupsPerK = 128;
  sparse_conv_wmma_f32<<<dim3(K * groupsPerK), 256, 0, stream>>>(
      feat, weight, in_idx, out_idx, out, R, groupsPerK);
}